// NF4WeightQuantizer_72249939853554
// MI455X (gfx1250) — compile-verified
//
#include <hip/hip_runtime.h>
#include <stdint.h>

// NF4 quantize-dequantize, 8192x8192 fp32, block=64.
// Memory-bound: 512 MB traffic -> ~22 us floor at 23.3 TB/s.
// CDNA5 paths: global_load_async_to_lds_b128 depth-2 prefetch pipeline
// (ASYNCcnt), ds_load_b128, ds-permute codebook gather, scalar-float expf,
// NT loads/stores (streamed data > 192 MB L2).

#define TPB  256   // 8 wave32 per workgroup
#define ITER 16    // float4 iterations per workgroup

typedef float v4f __attribute__((ext_vector_type(4)));

__device__ __constant__ float NF4_TAB[16] = {
    -1.0f, -0.6961928f, -0.52507305f, -0.39491749f, -0.28444138f,
    -0.18477343f, -0.09105004f, 0.0f, 0.0795803f, 0.1609302f,
    0.2461123f, 0.33791524f, 0.44070983f, 0.562617f, 0.72295684f, 1.0f};

// Async copy engine: copies 16 B/lane from global address base + voff to LDS offset lds_off, ASYNCcnt.
__device__ __forceinline__ void async_load_b128_to_lds(uint32_t lds_off,
                                                       uint32_t voff,
                                                       const float* base) {
  asm volatile("global_load_async_to_lds_b128 %0, %1, %2 th:TH_LOAD_NT"
               :: "v"(lds_off), "v"(voff), "s"(base)
               : "memory");
}

// Nearest NF4 level via branchless 4-level binary search over the 15
// midpoints (== searchsorted side='left' of the reference), then gather the
// level value from the codebook resident in lanes [0..15]/[16..31] via
// ds_bpermute (__shfl width 16). ~27 VALU/element vs ~50 for a linear scan.
__device__ __forceinline__ float nf4_qd(float x, float rs, float s, float tab) {
  const float M0 = -0.8480964f,  M1 = -0.61063293f, M2 = -0.45999527f,
              M3 = -0.33967944f, M4 = -0.23460741f, M5 = -0.13791174f,
              M6 = -0.04552502f, M7 =  0.03979015f, M8 =  0.12025525f,
              M9 =  0.20352125f, M10 = 0.29201377f, M11 = 0.38931254f,
              M12 = 0.50166342f, M13 = 0.64278692f, M14 = 0.86147842f;
  float xn = __builtin_amdgcn_fmed3f(x * rs, -1.0f, 1.0f);  // clamp in 1 op

  bool  b3 = xn > M7;                       // idx bit 3
  float m2 = b3 ? M11 : M3;
  bool  b2 = xn > m2;                       // idx bit 2
  float t  = b3 ? M13 : M5;
  float u  = b3 ? M9  : M1;
  float m1 = b2 ? t : u;
  bool  b1 = xn > m1;                       // idx bit 1
  float a  = b1 ? M2  : M0;                 // even midpoints M[8b3+4b2+2b1]
  float b  = b1 ? M6  : M4;
  float c  = b1 ? M10 : M8;
  float d  = b1 ? M14 : M12;
  float e  = b2 ? b : a;
  float f  = b2 ? d : c;
  float m0 = b3 ? f : e;
  bool  b0 = xn > m0;                       // idx bit 0

  int idx = (b3 ? 8 : 0) | (b2 ? 4 : 0) | (b1 ? 2 : 0) | (b0 ? 1 : 0);
  return __shfl(tab, idx, 16) * s;
}

__global__ void __launch_bounds__(TPB)
nf4_qdq_kernel(const float* __restrict__ w, const float* __restrict__ log_scale,
               float* __restrict__ out, int n4) {
  __shared__ v4f buf[4][TPB];  // 16 KB, 4-deep ring for distance-2 prefetch
  const int tid = threadIdx.x;
  const float es  = expf(log_scale[0]);     // uniform -> scalar-float ops
  const float tab = NF4_TAB[tid & 15];      // codebook resident per half-wave

  // addrspace(3) offset = low 32 bits of generic pointer; ring stride 4096 B
  const uint32_t lds_base = (uint32_t)(uintptr_t)&buf[0][tid];

  const long long base4 = (long long)blockIdx.x * (TPB * ITER);
  // Full tile -> depth-2 pipeline is hazard-free; partial tail workgroup
  // falls back to fully-serialized waits (always correct).
  const bool full = (base4 + (long long)TPB * ITER) <= (long long)n4;

  // Prologue: prefetch iterations 0 and 1.
  {
    long long g0 = base4 + tid;
    if (g0 < n4) async_load_b128_to_lds(lds_base, (uint32_t)(g0 * 16), w);
    long long g1 = g0 + TPB;
    if (g1 < n4) async_load_b128_to_lds(lds_base + 4096u, (uint32_t)(g1 * 16), w);
  }

#pragma unroll 1
  for (int i = 0; i < ITER; ++i) {
    long long g = base4 + (long long)i * TPB + tid;

    // Retire the oldest async load (loads complete in order); keep the
    // distance-2 prefetch in flight. Serialize on the tail workgroup.
    if (full && (i < ITER - 1))
      asm volatile("s_wait_asynccnt 0x1" ::: "memory");
    else
      asm volatile("s_wait_asynccnt 0x0" ::: "memory");

    v4f v = (g < n4) ? buf[i & 3][tid] : (v4f)0.0f;   // ds_load_b128

    // LDS read must land before this ring slot can be re-targeted by a
    // future async load (WAR vs. the copy engine).
    asm volatile("s_wait_dscnt 0x0" ::: "memory");

    // Prefetch i+2 into the next ring slot; overlaps compute + store.
    if (i + 2 < ITER) {
      long long gn = g + 2 * TPB;
      if (gn < n4)
        async_load_b128_to_lds(lds_base + ((uint32_t)((i + 2) & 3) << 12),
                               (uint32_t)(gn * 16), w);
    }

    if (g >= n4) continue;

    // Block absmax: per-lane max of 4, then xor-reduce across the 16-lane
    // half-wave (masks 1,2,4,8 never cross the 64-float block boundary).
    float ax = fmaxf(fmaxf(fabsf(v.x), fabsf(v.y)),
                     fmaxf(fabsf(v.z), fabsf(v.w)));
    ax = fmaxf(ax, __shfl_xor(ax, 1, 32));
    ax = fmaxf(ax, __shfl_xor(ax, 2, 32));
    ax = fmaxf(ax, __shfl_xor(ax, 4, 32));
    ax = fmaxf(ax, __shfl_xor(ax, 8, 32));

    float s  = fmaxf(ax * es, 1e-6f);
    float rs = 1.0f / s;

    v4f r;
    r.x = nf4_qd(v.x, rs, s, tab);
    r.y = nf4_qd(v.y, rs, s, tab);
    r.z = nf4_qd(v.z, rs, s, tab);
    r.w = nf4_qd(v.w, rs, s, tab);

    __builtin_nontemporal_store(r, reinterpret_cast<v4f*>(out) + g);
  }
}

extern "C" void kernel_launch(void* const* d_in, const int* in_sizes, int n_in,
                              void* d_out, int out_size, void* d_ws, size_t ws_size,
                              hipStream_t stream) {
  (void)n_in; (void)out_size; (void)d_ws; (void)ws_size;
  const float* w  = (const float*)d_in[0];
  const float* ls = (const float*)d_in[1];
  float* out = (float*)d_out;

  int n  = in_sizes[0];          // 8192*8192
  int n4 = n >> 2;               // float4 count (in_f % 64 == 0 guaranteed)
  int per_wg = TPB * ITER;
  int grid = (n4 + per_wg - 1) / per_wg;   // 4096 WGs for the 8192^2 case

  nf4_qdq_kernel<<<dim3(grid), dim3(TPB), 0, stream>>>(w, ls, out, n4);
}